// LSTMTransformerModel2_9053791060064
// MI455X (gfx1250) — compile-verified
//
#include <hip/hip_runtime.h>
#include <math.h>

#define NB 5
#define HID 1024
#define IN 42

typedef __attribute__((ext_vector_type(2))) float v2f;
typedef __attribute__((ext_vector_type(8))) float v8f;

// ---------------------------------------------------------------------------
// Input preparation: exactly _prepare_input from the reference. Tiny (133x2
// floats) -> one thread does it serially; negligible vs 112MB of weights.
// ---------------------------------------------------------------------------
__global__ void prep_kernel(const float* __restrict__ x, float* __restrict__ xb) {
    if (threadIdx.x != 0 || blockIdx.x != 0) return;

    float left[21][2], right[21][2];
    for (int i = 0; i < 21; ++i) {
        left[i][0]  = x[(91 + i) * 2 + 0];
        left[i][1]  = x[(91 + i) * 2 + 1];
        right[i][0] = x[(112 + i) * 2 + 0];
        right[i][1] = x[(112 + i) * 2 + 1];
    }
    float sbx = x[0], sby = x[1];
    float slx = x[100 * 2], sly = x[100 * 2 + 1];
    float srx = x[121 * 2], sry = x[121 * 2 + 1];

    float lmin0 = left[0][0], lmax0 = left[0][0], lmin1 = left[0][1], lmax1 = left[0][1];
    float rmin0 = right[0][0], rmax0 = right[0][0], rmin1 = right[0][1], rmax1 = right[0][1];
    for (int i = 1; i < 21; ++i) {
        lmin0 = fminf(lmin0, left[i][0]);  lmax0 = fmaxf(lmax0, left[i][0]);
        lmin1 = fminf(lmin1, left[i][1]);  lmax1 = fmaxf(lmax1, left[i][1]);
        rmin0 = fminf(rmin0, right[i][0]); rmax0 = fmaxf(rmax0, right[i][0]);
        rmin1 = fminf(rmin1, right[i][1]); rmax1 = fmaxf(rmax1, right[i][1]);
    }
    float w_l = lmax0 - lmin0, h_l = lmax1 - lmin1;
    float w_r = rmax0 - rmin0, h_r = rmax1 - rmin1;
    float w_b = x[5 * 2] - x[6 * 2];
    float h_b = 4.0f * w_b;

    bool cond_l = (w_l != 0.0f) && (h_l != 0.0f);
    bool cond_r = (w_r != 0.0f) && (h_r != 0.0f);
    bool cond_b = (w_b != 0.0f) && (h_b != 0.0f);
    float dl0 = cond_l ? w_l : 1.0f, dl1 = cond_l ? h_l : 1.0f;
    float dr0 = cond_r ? w_r : 1.0f, dr1 = cond_r ? h_r : 1.0f;
    float wbb = cond_b ? w_b : 1.0f, hbb = cond_b ? h_b : 1.0f;

    float feat[105][2];
    for (int i = 0; i < 21; ++i) {
        // left normalized
        feat[i][0]      = (left[i][0] - slx) / dl0;
        feat[i][1]      = (left[i][1] - sly) / dl1;
        // right normalized
        feat[21 + i][0] = (right[i][0] - srx) / dr0;
        feat[21 + i][1] = (right[i][1] - sry) / dr1;
        // f2l = (left - src_body) / [wb, 1]
        feat[42 + i][0] = (left[i][0] - sbx) / wbb;
        feat[42 + i][1] = (left[i][1] - sby);
        // f2r = (right - src_body) / [1, hb]
        feat[63 + i][0] = (right[i][0] - sbx);
        feat[63 + i][1] = (right[i][1] - sby) / hbb;
        // h2h = (left - right) / [wb, hb]
        feat[84 + i][0] = (left[i][0] - right[i][0]) / wbb;
        feat[84 + i][1] = (left[i][1] - right[i][1]) / hbb;
    }
    for (int k = 0; k < 210; ++k)
        xb[k] = feat[k >> 1][k & 1];
}

__device__ __forceinline__ float sigmoidf_(float v) { return 1.0f / (1.0f + expf(-v)); }

// ---------------------------------------------------------------------------
// LSTM cell 0: h,c start at 0 so g = W_ih0 @ x + b_ih0 + b_hh0 (K=42, tiny).
// One thread per (branch, hidden) output.
// ---------------------------------------------------------------------------
__global__ void lstm0_kernel(const float* __restrict__ Wih0,
                             const float* __restrict__ bih0,
                             const float* __restrict__ bhh0,
                             const float* __restrict__ xb,
                             float* __restrict__ h0) {
    int j = blockIdx.x * blockDim.x + threadIdx.x;  // 0..1023
    int b = blockIdx.y;                             // 0..4
    if (j >= HID) return;
    const float* xv = xb + b * IN;
    float g[4];
#pragma unroll
    for (int gt = 0; gt < 4; ++gt) {
        long row = (long)b * 4 * HID + gt * HID + j;
        const float* wr = Wih0 + row * IN;
        float s = bih0[row] + bhh0[row];
#pragma unroll
        for (int k = 0; k < IN; ++k) s += wr[k] * xv[k];
        g[gt] = s;
    }
    float c2 = sigmoidf_(g[0]) * tanhf(g[2]);     // c=0: forget path vanishes
    h0[b * HID + j] = sigmoidf_(g[3]) * tanhf(c2);
}

// ---------------------------------------------------------------------------
// Gate nonlinearity for cell 1 (g1 precomputed by the WMMA GEMV).
// Writes directly into ctx (branch-major 5*1024) since softmax(1-elt) == 1.
// ---------------------------------------------------------------------------
__global__ void gates1_kernel(const float* __restrict__ g1,
                              const float* __restrict__ bih1,
                              const float* __restrict__ bhh1,
                              float* __restrict__ ctx) {
    int j = blockIdx.x * blockDim.x + threadIdx.x;
    int b = blockIdx.y;
    if (j >= HID) return;
    long base = (long)b * 4 * HID;
    float gi = g1[base + j]            + bih1[base + j]            + bhh1[base + j];
    float gf = g1[base + HID + j]      + bih1[base + HID + j]      + bhh1[base + HID + j];
    float gg = g1[base + 2 * HID + j]  + bih1[base + 2 * HID + j]  + bhh1[base + 2 * HID + j];
    float go = g1[base + 3 * HID + j]  + bih1[base + 3 * HID + j]  + bhh1[base + 3 * HID + j];
    (void)gf;  // c = 0 -> sigmoid(f)*c == 0
    float c2 = sigmoidf_(gi) * tanhf(gg);
    ctx[b * HID + j] = sigmoidf_(go) * tanhf(c2);
}

// ---------------------------------------------------------------------------
// f32 WMMA GEMV: y[0:nrows] = W(nrows x K, row-major) @ x + bias.
// One wave per 16-row tile, marching K in steps of 4 with
// V_WMMA_F32_16X16X4_F32. A-tile ISA layout: lanes 0-15 carry K=k,k+1 of rows
// 0..15, lanes 16-31 carry K=k+2,k+3 -> one float2 load per lane per step
// (the 256B/WMMA weight stream is the whole bandwidth story). B is x[k..k+3]
// replicated across all 16 columns, so every column of C equals the y chunk.
// ---------------------------------------------------------------------------
__global__ __launch_bounds__(32) void wmma_gemv_kernel(
    const float* __restrict__ W, const float* __restrict__ xv,
    const float* __restrict__ bias, float* __restrict__ y,
    int nrows, int K, long Wstride, int xstride, int ystride) {
    int b = blockIdx.y;
    const float* Wb = W + (long)b * Wstride;
    const float* xp = xv + (long)b * xstride;
    float* yp = y + (long)b * ystride;

    int lane = threadIdx.x & 31;
    int half = lane >> 4;       // 0: K=k,k+1   1: K=k+2,k+3
    int m    = lane & 15;       // row within tile
    int row0 = blockIdx.x * 16;
    int row  = row0 + m;
    int rld  = row < nrows ? row : nrows - 1;   // clamp tail loads in-bounds
    const float* Wrow = Wb + (long)rld * K;

    v8f acc = {};
#pragma unroll 8
    for (int k = 0; k < K; k += 4) {
        int kk = k + 2 * half;
        float2 aw = *(const float2*)(Wrow + kk);   // 8B aligned: K%4==0
        float2 bx = *(const float2*)(xp + kk);     // broadcast chunk (cached)
        v2f a = { aw.x, aw.y };
        v2f bb = { bx.x, bx.y };
        acc = __builtin_amdgcn_wmma_f32_16x16x4_f32(
            /*neg_a=*/false, a, /*neg_b=*/false, bb,
            /*c_mod=*/(short)0, acc, /*reuse_a=*/false, /*reuse_b=*/false);
    }
    // C layout: lanes 0-15 hold M=0..7 in acc[0..7], lanes 16-31 hold M=8..15.
    // All columns identical -> lane with (m>>3)==half owns y[row0+m] in acc[m&7].
    if (((m >> 3) == half) && row < nrows) {
        float v;
        switch (m & 7) {
            case 0: v = acc[0]; break; case 1: v = acc[1]; break;
            case 2: v = acc[2]; break; case 3: v = acc[3]; break;
            case 4: v = acc[4]; break; case 5: v = acc[5]; break;
            case 6: v = acc[6]; break; default: v = acc[7]; break;
        }
        if (bias) v += bias[row];
        yp[row] = v;
    }
}

extern "C" void kernel_launch(void* const* d_in, const int* in_sizes, int n_in,
                              void* d_out, int out_size, void* d_ws, size_t ws_size,
                              hipStream_t stream) {
    const float* x      = (const float*)d_in[0];
    const float* W_ih0  = (const float*)d_in[1];
    // d_in[2] = W_hh0 : multiplied by h=0, never read
    const float* b_ih0  = (const float*)d_in[3];
    const float* b_hh0  = (const float*)d_in[4];
    const float* W_ih1  = (const float*)d_in[5];
    // d_in[6] = W_hh1 : multiplied by h=0, never read
    const float* b_ih1  = (const float*)d_in[7];
    const float* b_hh1  = (const float*)d_in[8];
    // d_in[9..12] = Wa, ba, Ws, bs : softmax over 1 element == 1.0 exactly,
    // so the attention weight never affects the output -> skipped bit-exactly.
    const float* fuse_w = (const float*)d_in[13];
    const float* fuse_b = (const float*)d_in[14];
    const float* fc_w   = (const float*)d_in[15];
    const float* fc_b   = (const float*)d_in[16];
    float* out = (float*)d_out;

    float* ws    = (float*)d_ws;
    float* xb    = ws;                        // 210   (pad to 256)
    float* h0    = ws + 256;                  // 5120
    float* g1    = ws + 256 + 5120;           // 20480
    float* ctx   = ws + 256 + 5120 + 20480;   // 5120
    float* fused = ws + 256 + 5120 + 20480 + 5120;  // 1024

    // 1) feature extraction -> xb (5 x 42)
    prep_kernel<<<1, 32, 0, stream>>>(x, xb);

    // 2) cell 0 (K=42, 3.4MB): plain VALU dot products
    lstm0_kernel<<<dim3(4, NB), 256, 0, stream>>>(W_ih0, b_ih0, b_hh0, xb, h0);

    // 3) cell 1 GEMV (dominant: 84MB of f32 weights) via f32 WMMA
    wmma_gemv_kernel<<<dim3(4 * HID / 16, NB), 32, 0, stream>>>(
        W_ih1, h0, nullptr, g1, 4 * HID, HID,
        (long)4 * HID * HID, HID, 4 * HID);

    // 4) gates -> ctx (softmax weight == 1)
    gates1_kernel<<<dim3(4, NB), 256, 0, stream>>>(g1, b_ih1, b_hh1, ctx);

    // 5) fused = fuse_w (1024 x 5120) @ ctx + fuse_b   (21MB)
    wmma_gemv_kernel<<<dim3(HID / 16, 1), 32, 0, stream>>>(
        fuse_w, ctx, fuse_b, fused, HID, NB * HID, 0, 0, 0);

    // 6) logits = fc_w (1000 x 1024) @ fused + fc_b    (4MB, 63 tiles w/ tail)
    wmma_gemv_kernel<<<dim3((1000 + 15) / 16, 1), 32, 0, stream>>>(
        fc_w, fused, fc_b, out, 1000, HID, 0, 0, 0);

    (void)in_sizes; (void)n_in; (void)out_size; (void)ws_size;
}